// SkipGramModel_78847009620712
// MI455X (gfx1250) — compile-verified
//
#include <hip/hip_runtime.h>
#include <math.h>

// -------- problem constants (match reference) --------
#define VOCAB  1000000
#define DIM    128
#define BATCH  65536
#define NEG    5

#define TPB    256                 // 8 wave32 per block
#define WAVES_PER_BLOCK (TPB / 32)
#define GROUPS (BATCH / 16)        // 4096 groups of 16 batch elements
#define BLOCKS (GROUPS / WAVES_PER_BLOCK)   // 512

typedef __attribute__((ext_vector_type(2))) float v2f;
typedef __attribute__((ext_vector_type(8))) float v8f;

// Fast stable logsigmoid: logsig(x) = min(x,0) - log(1 + exp(-|x|)).
// Uses hardware v_exp_f32 / v_log_f32; no cancellation (log argument in [1,2]).
__device__ __forceinline__ float fast_logsigmoid(float x) {
    return fminf(x, 0.0f) - __logf(1.0f + __expf(-fabsf(x)));
}

// Stage one 64-float (256B contiguous) row slice into registers:
// 16 back-to-back global_load_b128, single wait at first use.
__device__ __forceinline__ void load_row(float4 dst[16], const float* __restrict__ base) {
    const float4* p = (const float4*)base;
#pragma unroll
    for (int t = 0; t < 16; ++t) dst[t] = p[t];
}

// Per-lane partial dots + WMMA row-sum: D = A(16x4) * ones(4x16) =>
// D[m,n] = full 128-dim dot for batch row m (replicated across all columns n).
// A f32 layout (ISA 7.12.2): lane m   -> VGPR0=K0, VGPR1=K1
//                            lane m+16-> VGPR0=K2, VGPR1=K3
// After the WMMA, lane l<16 holds rows 0..7 in C[0..7]; lane l>=16 holds 8..15.
// Distribute: lanes 0..7 take C[l] (rows 0..7), lanes 16..23 take C[l-16]
// (rows 8..15); remaining lanes contribute 0. One logsigmoid per lane.
__device__ __forceinline__ float row_score(const float4 v[16], const float4 a[16],
                                           v2f ones, float sign,
                                           int idx, bool valid) {
    float a0 = 0.0f, a1 = 0.0f;
#pragma unroll
    for (int t = 0; t < 8; ++t) {
        a0 += a[t].x * v[t].x + a[t].y * v[t].y + a[t].z * v[t].z + a[t].w * v[t].w;
    }
#pragma unroll
    for (int t = 8; t < 16; ++t) {
        a1 += a[t].x * v[t].x + a[t].y * v[t].y + a[t].z * v[t].z + a[t].w * v[t].w;
    }
    v2f A; A.x = a0; A.y = a1;
    v8f C = {};
    C = __builtin_amdgcn_wmma_f32_16x16x4_f32(
            /*neg_a=*/false, A, /*neg_b=*/false, ones,
            /*c_mod=*/(short)0, C, /*reuse_a=*/false, /*reuse_b=*/false);

    // Register-indexed pick via short cndmask chain (idx = lane & 7).
    float x = C[0];
#pragma unroll
    for (int r = 1; r < 8; ++r) x = (idx == r) ? C[r] : x;

    float ls = fast_logsigmoid(sign * x);
    return valid ? ls : 0.0f;
}

__global__ __launch_bounds__(TPB)
void sg_loss_partials(const int*   __restrict__ pos_w,
                      const int*   __restrict__ pos_v,
                      const int*   __restrict__ neg_v,
                      const float* __restrict__ w_table,
                      const float* __restrict__ v_table,
                      float*       __restrict__ partials)
{
    const int tid  = threadIdx.x;
    const int lane = tid & 31;
    const int wave = tid >> 5;
    const int m    = lane & 15;         // A-matrix row this lane feeds
    const int hx   = (lane >> 4) * 64;  // which half of DIM this lane covers
    const int group = blockIdx.x * WAVES_PER_BLOCK + wave;  // 0..4095
    const int b = group * 16 + m;       // batch element for row m

    const int  sidx   = lane & 7;            // which C-register this lane extracts
    const bool svalid = (lane & 8) == 0;     // lanes 0-7 and 16-23 carry results

    // Issue all index gathers up front so row pointers are ready early.
    const int iv  = pos_v[b];
    const int iw  = pos_w[b];
    int in[NEG];
#pragma unroll
    for (int j = 0; j < NEG; ++j) in[j] = neg_v[b * NEG + j];

    const float* vbase = v_table + (size_t)iv * DIM + hx;
    const float* wbase = w_table + (size_t)iw * DIM + hx;

    // This lane's 64-float slice of emb_v, kept in VGPRs for all 6 rows.
    float4 vb[16];
    load_row(vb, vbase);

    v2f ones; ones.x = 1.0f; ones.y = 1.0f;

    // Software pipeline: double-buffer A-rows so the next row's 16 loads are
    // in flight while the current row's FMAs + WMMA + logsigmoid execute.
    float4 bufA[16], bufB[16];
    float s = 0.0f;

    load_row(bufA, wbase);                                        // pos row
    load_row(bufB, v_table + (size_t)in[0] * DIM + hx);           // neg0
    s += row_score(vb, bufA, ones,  1.0f, sidx, svalid);          // pos
    load_row(bufA, v_table + (size_t)in[1] * DIM + hx);           // neg1
    s += row_score(vb, bufB, ones, -1.0f, sidx, svalid);          // neg0
    load_row(bufB, v_table + (size_t)in[2] * DIM + hx);           // neg2
    s += row_score(vb, bufA, ones, -1.0f, sidx, svalid);          // neg1
    load_row(bufA, v_table + (size_t)in[3] * DIM + hx);           // neg3
    s += row_score(vb, bufB, ones, -1.0f, sidx, svalid);          // neg2
    load_row(bufB, v_table + (size_t)in[4] * DIM + hx);           // neg4
    s += row_score(vb, bufA, ones, -1.0f, sidx, svalid);          // neg3
    s += row_score(vb, bufB, ones, -1.0f, sidx, svalid);          // neg4

    // Full wave reduction (fixed tree order -> deterministic): lanes 0..7 and
    // 16..23 hold the 16 distinct per-element score sums, others hold 0.
#pragma unroll
    for (int off = 16; off > 0; off >>= 1) s += __shfl_xor(s, off, 32);

    __shared__ float lds[WAVES_PER_BLOCK];
    if (lane == 0) lds[wave] = s;
    __syncthreads();
    if (tid == 0) {
        float bp = 0.0f;
#pragma unroll
        for (int i = 0; i < WAVES_PER_BLOCK; ++i) bp += lds[i];
        partials[blockIdx.x] = bp;   // deterministic fixed-order sum
    }
}

// Deterministic tree reduction of the 512 block partials; loss = -(sum).
__global__ __launch_bounds__(256)
void sg_loss_reduce(const float* __restrict__ partials,
                    float*       __restrict__ out)
{
    __shared__ float lds[256];
    const int t = threadIdx.x;
    lds[t] = partials[t] + partials[t + 256];
    __syncthreads();
    for (int stride = 128; stride > 0; stride >>= 1) {
        if (t < stride) lds[t] += lds[t + stride];
        __syncthreads();
    }
    if (t == 0) out[0] = -lds[0];
}

extern "C" void kernel_launch(void* const* d_in, const int* in_sizes, int n_in,
                              void* d_out, int out_size, void* d_ws, size_t ws_size,
                              hipStream_t stream) {
    const int*   pos_w   = (const int*)d_in[0];   // [BATCH]
    const int*   pos_v   = (const int*)d_in[1];   // [BATCH]
    const int*   neg_v   = (const int*)d_in[2];   // [BATCH, NEG]
    const float* w_table = (const float*)d_in[3]; // [VOCAB, DIM]
    const float* v_table = (const float*)d_in[4]; // [VOCAB, DIM]
    float* out = (float*)d_out;
    float* partials = (float*)d_ws;               // BLOCKS floats of scratch

    sg_loss_partials<<<BLOCKS, TPB, 0, stream>>>(pos_w, pos_v, neg_v,
                                                 w_table, v_table, partials);
    sg_loss_reduce<<<1, 256, 0, stream>>>(partials, out);
}